// LGCN_Encoder_74783970558254
// MI455X (gfx1250) — compile-verified
//
#include <hip/hip_runtime.h>

#define USER_NUM 50000
#define ITEM_NUM 50000
#define N_TOTAL  (USER_NUM + ITEM_NUM)   // 100000; % 16 == 0
#define FREQ     512
#define EMB      64
#define MAX_SPLIT 40
#define PROJ_ELEMS (FREQ * EMB)          // 32768 floats = 128 KB

typedef __attribute__((ext_vector_type(2))) float v2f;
typedef __attribute__((ext_vector_type(8))) float v8f;

// D(16x16,f32) = A(16x4,f32) x B(4x16,f32) + C   -> v_wmma_f32_16x16x4_f32
__device__ __forceinline__ v8f wmma_f32(v2f a, v2f b, v8f c) {
  return __builtin_amdgcn_wmma_f32_16x16x4_f32(
      /*neg_a=*/false, a, /*neg_b=*/false, b,
      /*c_mod=*/(short)0, c, /*reuse_a=*/false, /*reuse_b=*/false);
}

// ---------------------------------------------------------------------------
// Kernel 1 helper: accumulate over [beg, end) (multiples of 4) with a single
// rebased ego pointer (no per-iteration user/item select).
// ---------------------------------------------------------------------------
__device__ __forceinline__ void accum_segment(const float* __restrict__ v,
                                              const float* __restrict__ ego,
                                              int beg, int end,
                                              int m, int kh, int f0,
                                              v8f& acc0, v8f& acc1,
                                              v8f& acc2, v8f& acc3) {
  #pragma unroll 4
  for (int n0 = beg; n0 < end; n0 += 4) {
    const int r0 = n0 + 2 * kh;            // this half-wave's K pair rows
    // A[m,k] = v[n0+k, f0+m] : contiguous in f across lanes; streamed once.
    const float* __restrict__ av = v + (size_t)r0 * FREQ + f0 + m;
    v2f a;
    a.x = __builtin_nontemporal_load(av);
    a.y = __builtin_nontemporal_load(av + FREQ);
    // B[k,j] = ego[n0+k, j] : lane-contiguous; L2-resident (reused 32x).
    const float* __restrict__ e0 = ego + (size_t)r0 * EMB + m;
    const float* __restrict__ e1 = e0 + EMB;
    v2f b;
    b.x = e0[ 0]; b.y = e1[ 0]; acc0 = wmma_f32(a, b, acc0);
    b.x = e0[16]; b.y = e1[16]; acc1 = wmma_f32(a, b, acc1);
    b.x = e0[32]; b.y = e1[32]; acc2 = wmma_f32(a, b, acc2);
    b.x = e0[48]; b.y = e1[48]; acc3 = wmma_f32(a, b, acc3);
  }
}

// ---------------------------------------------------------------------------
// Kernel 1: partials[sp][f][d] = sum_{n in slice sp} v[n,f] * ego[n,d]
// One wave = one 16-wide f tile x all 64 d, over a slice of N.
// ---------------------------------------------------------------------------
__global__ void __launch_bounds__(128)
proj_partial_kernel(const float* __restrict__ user_emb,
                    const float* __restrict__ item_emb,
                    const float* __restrict__ v,
                    float* __restrict__ partials,
                    int chunk) {
  const int lane  = threadIdx.x & 31;
  const int wave  = threadIdx.x >> 5;
  const int fTile = blockIdx.x * 4 + wave;   // 0..31 (always valid)
  const int sp    = blockIdx.y;
  const int m     = lane & 15;
  const int kh    = lane >> 4;
  const int f0    = fTile * 16;

  int nBeg = sp * chunk;                     // chunk % 4 == 0
  int nEnd = nBeg + chunk;
  if (nEnd > N_TOTAL) nEnd = N_TOTAL;        // N_TOTAL % 4 == 0

  v8f acc0 = {0,0,0,0,0,0,0,0};
  v8f acc1 = {0,0,0,0,0,0,0,0};
  v8f acc2 = {0,0,0,0,0,0,0,0};
  v8f acc3 = {0,0,0,0,0,0,0,0};

  // Split at the user/item boundary (USER_NUM % 4 == 0): select hoisted out.
  const int uEnd = (nEnd < USER_NUM) ? nEnd : USER_NUM;
  const int iBeg = (nBeg > USER_NUM) ? nBeg : USER_NUM;
  if (nBeg < uEnd)
    accum_segment(v, user_emb, nBeg, uEnd, m, kh, f0, acc0, acc1, acc2, acc3);
  if (iBeg < nEnd)
    accum_segment(v, item_emb - (size_t)USER_NUM * EMB, iBeg, nEnd,
                  m, kh, f0, acc0, acc1, acc2, acc3);

  // C/D layout: VGPR r, lane<16 -> M=r, lane>=16 -> M=r+8, Ncol = lane&15
  float* __restrict__ dst = partials + (size_t)sp * PROJ_ELEMS;
  #pragma unroll
  for (int r = 0; r < 8; ++r) {
    const int fr = f0 + r + 8 * kh;
    dst[fr * EMB +  0 + m] = acc0[r];
    dst[fr * EMB + 16 + m] = acc1[r];
    dst[fr * EMB + 32 + m] = acc2[r];
    dst[fr * EMB + 48 + m] = acc3[r];
  }
}

// ---------------------------------------------------------------------------
// Kernel 2: projS[f,d] = filt[f] * sum_sp partials[sp][f][d]   (deterministic)
// ---------------------------------------------------------------------------
__global__ void reduce_scale_kernel(const float* __restrict__ partials,
                                    const float* __restrict__ filt,
                                    float* __restrict__ projS,
                                    int split_cnt) {
  const int i = blockIdx.x * blockDim.x + threadIdx.x;   // f*64 + d
  if (i < PROJ_ELEMS) {
    float s = 0.f;
    for (int p = 0; p < split_cnt; ++p)
      s += partials[(size_t)p * PROJ_ELEMS + i];
    projS[i] = s * filt[i >> 6];
  }
}

// ---------------------------------------------------------------------------
// Kernel 3: out[n,d] = 0.5 * (ego[n,d] + sum_f v[n,f] * projS[f,d])
// One wave = one 16-row n tile x all 64 d.
// ---------------------------------------------------------------------------
__global__ void __launch_bounds__(128)
filter_combine_kernel(const float* __restrict__ user_emb,
                      const float* __restrict__ item_emb,
                      const float* __restrict__ v,
                      const float* __restrict__ projS,
                      float* __restrict__ out) {
  const int lane  = threadIdx.x & 31;
  const int wave  = threadIdx.x >> 5;
  const int nTile = blockIdx.x * 4 + wave;
  if (nTile * 16 >= N_TOTAL) return;         // wave-uniform guard
  const int m  = lane & 15;
  const int kh = lane >> 4;
  const int n0 = nTile * 16;

  v8f acc0 = {0,0,0,0,0,0,0,0};
  v8f acc1 = {0,0,0,0,0,0,0,0};
  v8f acc2 = {0,0,0,0,0,0,0,0};
  v8f acc3 = {0,0,0,0,0,0,0,0};

  // A[m,k] = v[n0+m, f0+k]: single b64 load per lane per step; each touched
  // cache line is fully consumed across consecutive k-steps of the same rows.
  const v2f* __restrict__ ap =
      reinterpret_cast<const v2f*>(v + (size_t)(n0 + m) * FREQ) + kh;

  #pragma unroll 4
  for (int f0 = 0; f0 < FREQ; f0 += 4) {
    v2f a = ap[f0 >> 1];                     // = v[n0+m, f0+2kh .. f0+2kh+1]
    // B[k,j] = projS[f0+k, j] (lane-contiguous, 128 KB L2-resident)
    const float* __restrict__ p0 = projS + (size_t)(f0 + 2 * kh) * EMB + m;
    const float* __restrict__ p1 = p0 + EMB;
    v2f b;
    b.x = p0[ 0]; b.y = p1[ 0]; acc0 = wmma_f32(a, b, acc0);
    b.x = p0[16]; b.y = p1[16]; acc1 = wmma_f32(a, b, acc1);
    b.x = p0[32]; b.y = p1[32]; acc2 = wmma_f32(a, b, acc2);
    b.x = p0[48]; b.y = p1[48]; acc3 = wmma_f32(a, b, acc3);
  }

  // Fused epilogue: out = (ego + filtered) * 0.5. Tile is entirely on one
  // side of the user/item boundary (USER_NUM % 16 == 0) -> one rebased base.
  const float* __restrict__ ego =
      (n0 < USER_NUM) ? user_emb : (item_emb - (size_t)USER_NUM * EMB);
  #pragma unroll
  for (int r = 0; r < 8; ++r) {
    const int row = n0 + r + 8 * kh;
    const float* __restrict__ er = ego + (size_t)row * EMB + m;
    float* __restrict__ orow = out + (size_t)row * EMB + m;
    __builtin_nontemporal_store((er[ 0] + acc0[r]) * 0.5f, orow +  0);
    __builtin_nontemporal_store((er[16] + acc1[r]) * 0.5f, orow + 16);
    __builtin_nontemporal_store((er[32] + acc2[r]) * 0.5f, orow + 32);
    __builtin_nontemporal_store((er[48] + acc3[r]) * 0.5f, orow + 48);
  }
}

// ---------------------------------------------------------------------------
extern "C" void kernel_launch(void* const* d_in, const int* in_sizes, int n_in,
                              void* d_out, int out_size, void* d_ws, size_t ws_size,
                              hipStream_t stream) {
  const float* user_emb = (const float*)d_in[0];   // [50000, 64]
  const float* item_emb = (const float*)d_in[1];   // [50000, 64]
  const float* v        = (const float*)d_in[2];   // [100000, 512]
  const float* filt     = (const float*)d_in[3];   // [512]
  // d_in[4] ("k") is unused by the reference.
  float* out = (float*)d_out;                      // [100000, 64]

  // Workspace: split_cnt partial slots + 1 slot for projS (128 KB each).
  const size_t slot = (size_t)PROJ_ELEMS * sizeof(float);
  int split_cnt = (int)(ws_size / slot) - 1;
  if (split_cnt > MAX_SPLIT) split_cnt = MAX_SPLIT;
  if (split_cnt < 1) split_cnt = 1;

  float* partials = (float*)d_ws;
  float* projS    = partials + (size_t)split_cnt * PROJ_ELEMS;

  // Per-split N slice, rounded up to a multiple of 4 (WMMA K step).
  int chunk = ((N_TOTAL + split_cnt - 1) / split_cnt + 3) & ~3;

  dim3 gridA(FREQ / 16 / 4, split_cnt);            // 8 x split blocks, 4 waves
  proj_partial_kernel<<<gridA, 128, 0, stream>>>(user_emb, item_emb, v,
                                                 partials, chunk);

  reduce_scale_kernel<<<(PROJ_ELEMS + 255) / 256, 256, 0, stream>>>(
      partials, filt, projS, split_cnt);

  const int nTiles = N_TOTAL / 16;                 // 6250
  filter_combine_kernel<<<(nTiles + 3) / 4, 128, 0, stream>>>(
      user_emb, item_emb, v, projS, out);
}